// FractalNeuralNetwork_31885837205972
// MI455X (gfx1250) — compile-verified
//
#include <hip/hip_runtime.h>

typedef __attribute__((ext_vector_type(2))) float v2f;
typedef __attribute__((ext_vector_type(8))) float v8f;

#define IMG_H 224
#define NIMG 64
#define PQ_THR 0.59275f

// ---------------------------------------------------------------------------
// Stage 1: per-patch binary counts -> per-image histograms (LDS -> global WS)
// WS layout (uint32): for box index bi, base off[bi]; per (dist d, image img):
//   entry = off + (d*64 + img)*(L+1);  bins [0..L-1] = hist, [L] = pq count
// ---------------------------------------------------------------------------
template <int B>
__global__ void fractal_count_kernel(const float* __restrict__ x,
                                     unsigned* __restrict__ gw, int off) {
  constexpr int L    = B * B;
  constexpr int ROWS = (IMG_H + B - 1) / B;
  constexpr int P    = ROWS * ROWS;
  constexpr int PAD  = (ROWS * B - IMG_H) / 2;  // top/left pad (ph//2)
  constexpr int CB   = (B - 1) / 2;
  constexpr int SH   = 3 * (L + 1);
  __shared__ unsigned sh[SH];

  const int tid = threadIdx.x;
  const int img = blockIdx.y;
  for (int i = tid; i < SH; i += blockDim.x) sh[i] = 0u;
  __syncthreads();

  const int p = blockIdx.x * blockDim.x + tid;
  if (p < P) {
    const int pr = p / ROWS, pc = p % ROWS;
    const float* xb = x + (size_t)img * IMG_H * IMG_H * 3;

    // center pixel of the (padded) patch; zero if it falls in padding
    float c0 = 0.f, c1 = 0.f, c2 = 0.f;
    {
      const int gy = pr * B + CB - PAD, gx = pc * B + CB - PAD;
      if (gy >= 0 && gy < IMG_H && gx >= 0 && gx < IMG_H) {
        const float* q = xb + ((size_t)gy * IMG_H + gx) * 3;
        c0 = q[0]; c1 = q[1]; c2 = q[2];
      }
    }

    const float thr  = (float)B;
    const float thr2 = thr * thr;  // exact for B in {3,5,7,9,11}
    int cc = 0, ce = 0, cm = 0;
    for (int i = 0; i < B; ++i) {
      const int gy = pr * B + i - PAD;
      const bool yok = (gy >= 0 && gy < IMG_H);
      const float* row = xb + (size_t)(yok ? gy : 0) * IMG_H * 3;
      if (yok && i + 1 < B && gy + 1 < IMG_H)
        __builtin_prefetch(row + (size_t)IMG_H * 3, 0, 1);  // global_prefetch_b8
      for (int j = 0; j < B; ++j) {
        const int gx = pc * B + j - PAD;
        const bool ok = yok && (gx >= 0) && (gx < IMG_H);
        float v0 = 0.f, v1 = 0.f, v2 = 0.f;
        if (ok) {
          const float* q = row + (size_t)gx * 3;
          v0 = q[0]; v1 = q[1]; v2 = q[2];
        }
        const float d0 = v0 - c0, d1 = v1 - c1, d2 = v2 - c2;
        const float a0 = fabsf(d0), a1 = fabsf(d1), a2 = fabsf(d2);
        cc += (fmaxf(fmaxf(a0, a1), a2) <= thr) ? 1 : 0;
        // sqrt(s) <= thr  <=>  s <= thr^2 (thr^2 exact, sqrt monotone/correct)
        ce += ((v0 * v0 + v1 * v1 + v2 * v2) <= thr2) ? 1 : 0;
        cm += ((a0 + a1 + a2) <= thr) ? 1 : 0;
      }
    }

    const float Lf = (float)L;
    const int cs[3] = {cc, ce, cm};
#pragma unroll
    for (int d = 0; d < 3; ++d) {
      const int c = cs[d];
      if (c < L) atomicAdd(&sh[d * (L + 1) + c], 1u);  // min(c,L-1), drop c==L
      if (((float)c / Lf) >= PQ_THR) atomicAdd(&sh[d * (L + 1) + L], 1u);
    }
  }
  __syncthreads();

  for (int i = tid; i < SH; i += blockDim.x) {
    const unsigned v = sh[i];
    if (v) {
      const int d = i / (L + 1), k = i % (L + 1);
      atomicAdd(&gw[(size_t)off + ((size_t)d * NIMG + img) * (L + 1) + k], v);
    }
  }
}

// ---------------------------------------------------------------------------
// Stage 2: histogram -> (fd, lac, pq) via V_WMMA_F32_16X16X4_F32.
//   D[64x16] = probs[64xL] x W[Lx16],  W cols: {1/n, n, n^2, 0...}, n = k+1.
// Block = 128 threads (4 waves); wave w handles images 16w..16w+15.
// A layout (16x4 f32): lane M = lane%16; VGPR0 = K={0|2}, VGPR1 = K={1|3}.
// D layout (16x16 f32): VGPR v -> M = v + 8*(lane>=16), N = lane%16.
// Branch-free K-loop: exact 1/(k+1) table precomputed in LDS (one IEEE div per
// k per block), clamp-indexed unconditional loads, cndmask masking, reciprocal
// of the row sum hoisted (rsum==0 -> inf -> 0*inf = NaN, matching ref 0/0).
// ---------------------------------------------------------------------------
__global__ void fractal_finalize_kernel(const unsigned* __restrict__ gw,
                                        float* __restrict__ out) {
  const int b_idx = blockIdx.x;  // 0..4 (box sizes 3,5,7,9,11)
  const int d     = blockIdx.y;  // 0..2 (cheb, eucl, manh)
  const int Ls[5]   = {9, 25, 49, 81, 121};
  const int offs[5] = {0, 1920, 6912, 16512, 32256};
  const int Ps[5]   = {75 * 75, 45 * 45, 32 * 32, 25 * 25, 21 * 21};
  const int L   = Ls[b_idx];
  const int off = offs[b_idx];
  const int P   = Ps[b_idx];

  const int tid  = threadIdx.x;  // 0..127
  const int wave = tid >> 5;     // 0..3
  const int lane = tid & 31;
  const int m    = lane & 15;    // A row / D column index
  const int hi   = lane >> 4;    // half-wave
  const int img  = wave * 16 + m;

  __shared__ float winv[121];    // exact 1/(k+1), built once per block
  __shared__ float tile[4][16][16];
  for (int k = tid; k < L; k += 128) winv[k] = 1.0f / (float)(k + 1);
  __syncthreads();

  const unsigned* hrow = gw + (size_t)off + ((size_t)d * NIMG + img) * (L + 1);

  float rsum = 0.f;
  for (int j = 0; j < L; ++j) rsum += (float)hrow[j];
  const float rinv = 1.0f / rsum;  // inf when rsum==0 -> NaN features (== ref 0/0)

  v8f acc = {};
  const int koff = hi * 2;
  for (int k0 = 0; k0 < L; k0 += 4) {
    const int ka0 = k0 + koff, ka1 = ka0 + 1;
    const bool in0 = (ka0 < L), in1 = (ka1 < L);
    const int ia0 = in0 ? ka0 : (L - 1);  // clamp -> unconditional loads
    const int ia1 = in1 ? ka1 : (L - 1);
    const float h0 = (float)hrow[ia0];
    const float h1 = (float)hrow[ia1];
    const float i0 = winv[ia0];           // ds_load, no division in loop
    const float i1 = winv[ia1];
    v2f a, bw;
    a.x = in0 ? h0 * rinv : 0.f;
    a.y = in1 ? h1 * rinv : 0.f;
    const float n0 = (float)(ka0 + 1), n1 = (float)(ka1 + 1);
    const float w0 = (m == 0) ? i0 : (m == 1) ? n0 : (m == 2) ? n0 * n0 : 0.f;
    const float w1 = (m == 0) ? i1 : (m == 1) ? n1 : (m == 2) ? n1 * n1 : 0.f;
    bw.x = in0 ? w0 : 0.f;
    bw.y = in1 ? w1 : 0.f;
    acc = __builtin_amdgcn_wmma_f32_16x16x4_f32(false, a, false, bw,
                                                (short)0, acc, false, false);
  }

#pragma unroll
  for (int v = 0; v < 8; ++v) tile[wave][v + hi * 8][m] = acc[v];
  __syncthreads();

  if (lane < 16) {
    const float fd  = tile[wave][m][0];
    const float mu1 = tile[wave][m][1];
    const float mu2 = tile[wave][m][2];
    const float lac = (mu2 - mu1 * mu1) / (mu1 * mu1);
    const float pq  = (float)hrow[L] / (float)P;
    float* o = out + (size_t)img * 45 + d * 15 + b_idx * 3;
    o[0] = fd; o[1] = lac; o[2] = pq;
  }
}

// ---------------------------------------------------------------------------
extern "C" void kernel_launch(void* const* d_in, const int* in_sizes, int n_in,
                              void* d_out, int out_size, void* d_ws, size_t ws_size,
                              hipStream_t stream) {
  const float* x = (const float*)d_in[0];
  unsigned* gw = (unsigned*)d_ws;
  float* out = (float*)d_out;

  // total WS: 3*64*Sum(L+1) = 55680 uints (~218 KB)
  hipMemsetAsync(d_ws, 0, (size_t)55680 * sizeof(unsigned), stream);

  const dim3 blk(256, 1, 1);
  fractal_count_kernel<3><<<dim3((75 * 75 + 255) / 256, NIMG), blk, 0, stream>>>(x, gw, 0);
  fractal_count_kernel<5><<<dim3((45 * 45 + 255) / 256, NIMG), blk, 0, stream>>>(x, gw, 1920);
  fractal_count_kernel<7><<<dim3((32 * 32 + 255) / 256, NIMG), blk, 0, stream>>>(x, gw, 6912);
  fractal_count_kernel<9><<<dim3((25 * 25 + 255) / 256, NIMG), blk, 0, stream>>>(x, gw, 16512);
  fractal_count_kernel<11><<<dim3((21 * 21 + 255) / 256, NIMG), blk, 0, stream>>>(x, gw, 32256);

  fractal_finalize_kernel<<<dim3(5, 3), dim3(128, 1, 1), 0, stream>>>(gw, out);
}